// LipschitzRNN_5617817223330
// MI455X (gfx1250) — compile-verified
//
#include <hip/hip_runtime.h>
#include <hip/hip_bf16.h>
#include <math.h>

#define NN 256
#define BSZ 256
#define TMAXX 512
#define STEPF 0.01f
#define YAF 0.001f
#define YWF 0.001f

// C matrix split: K in [0,128) lives in VGPR fragments, K in [128,256) in LDS.
#define CSTRIDE 132   // LDS row stride (floats) for C-half: bank = 4*row + col (conflict-free)

typedef __attribute__((ext_vector_type(2))) float v2f;
typedef __attribute__((ext_vector_type(8))) float v8f;

// f32 WMMA: D(16x16) = A(16x4) * B(4x16) + C   (V_WMMA_F32_16X16X4_F32)
static __device__ __forceinline__ v8f wmma4(v2f a, v2f b, v8f c) {
    return __builtin_amdgcn_wmma_f32_16x16x4_f32(
        false, a, false, b, (short)0, c, false, false);
}

// One-time prep: A = MA + (1-2bA)*MA^T - YA*I ; C = (1-bA+bW)*MW + (1-bA-bW)*MW^T - YW*I
__global__ __launch_bounds__(256) void lrnn_prep(
    const float* __restrict__ MA, const float* __restrict__ MW,
    const float* __restrict__ bAz, const float* __restrict__ bWz,
    float* __restrict__ A, float* __restrict__ C) {
    const int i = blockIdx.x;
    const int j = threadIdx.x;
    const float zA = bAz[0];
    const float zW = bWz[0];
    const float bA = 0.5f * expf(-zA * zA) + 0.5f;  // exp01
    const float bW = 0.5f * expf(-zW * zW) + 0.5f;
    const float ma_ij = MA[i * NN + j];
    const float ma_ji = MA[j * NN + i];
    const float mw_ij = MW[i * NN + j];
    const float mw_ji = MW[j * NN + i];
    const float d = (i == j) ? 1.0f : 0.0f;
    A[i * NN + j] = ma_ij + (1.0f - 2.0f * bA) * ma_ji - YAF * d;
    C[i * NN + j] = (1.0f - bA + bW) * mw_ij + (1.0f - bA - bW) * mw_ji - YWF * d;
}

// Persistent recurrence kernel: 16 workgroups x 512 threads (16 waves).
// Workgroup g owns batch columns [16g,16g+16). Wave w owns state rows [16w,16w+16).
__global__ __launch_bounds__(512) void lrnn_main(
    const float* __restrict__ X0, const float* __restrict__ A,
    const float* __restrict__ Cg, const float* __restrict__ by,
    float* __restrict__ out) {
    // X tile column-major with pad: Xs[col][row], stride 260 -> conflict-free ds_load_b64
    __shared__ float Xs[16][260];
    // C[:, 128:256] resident in LDS, padded stride 132
    __shared__ float Cs[NN * CSTRIDE];

    const int tid  = threadIdx.x;
    const int wav  = tid >> 5;       // 0..15 : row tile
    const int lane = tid & 31;
    const int lo   = lane & 15;      // batch col for B/C/D frags
    const int hi   = lane >> 4;      // half-wave select
    const int bc0  = blockIdx.x * 16;

    // Load X0 tile (transposed: Xs[col][row] = X0[b][row]) and emit t=0 slice.
    for (int idx = tid; idx < 16 * NN; idx += 512) {
        const int col = idx >> 8;    // 0..15
        const int row = idx & 255;
        const float v = X0[(size_t)(bc0 + col) * NN + row];
        Xs[col][row] = v;
        out[(size_t)(bc0 + col) * TMAXX * NN + row] = v;  // t = 0
    }
    // Stage C[:, 128:256) into LDS (coalesced).
    for (int idx = tid; idx < NN * 128; idx += 512) {
        const int row = idx >> 7;
        const int col = idx & 127;
        Cs[row * CSTRIDE + col] = Cg[row * NN + 128 + col];
    }

    // Persistent fragments (ISA 32-bit A-matrix 16x4 layout):
    // lane<16: row=16w+lane, K=4k+{0,1}; lane>=16: row=16w+lane-16, K=4k+{2,3}
    const int rowA = wav * 16 + lo;
    v2f aF[64];   // all of A's K range        -> 128 VGPRs
    v2f cF[32];   // C for K in [0,128)        ->  64 VGPRs
#pragma unroll
    for (int k = 0; k < 64; ++k)
        aF[k] = *(const v2f*)(A + (size_t)rowA * NN + 4 * k + 2 * hi);
#pragma unroll
    for (int k = 0; k < 32; ++k)
        cF[k] = *(const v2f*)(Cg + (size_t)rowA * NN + 4 * k + 2 * hi);

    // D layout: VGPR j holds rows M=j (lanes 0-15) and M=j+8 (lanes 16-31)
    const int rbase = wav * 16 + 8 * hi;
    float byF[8];
#pragma unroll
    for (int j = 0; j < 8; ++j) byF[j] = by[rbase + j];

    __syncthreads();

    float* outLane = out + (size_t)(bc0 + lo) * TMAXX * NN + rbase;
    const float* crow = &Cs[rowA * CSTRIDE];   // this wave-lane's C-half row

#pragma unroll 1
    for (int t = 1; t < TMAXX; ++t) {
        v8f accA = {0.f, 0.f, 0.f, 0.f, 0.f, 0.f, 0.f, 0.f};
        v8f accC = {0.f, 0.f, 0.f, 0.f, 0.f, 0.f, 0.f, 0.f};
        // K in [0,128): both matrices from registers
#pragma unroll
        for (int k = 0; k < 32; ++k) {
            const int kr = 4 * k + 2 * hi;
            const v2f b = *(const v2f*)&Xs[lo][kr];
            accA = wmma4(aF[k], b, accA);
            accC = wmma4(cF[k], b, accC);
        }
        // K in [128,256): A from registers, C from LDS
#pragma unroll
        for (int k = 32; k < 64; ++k) {
            const int kr = 4 * k + 2 * hi;
            const v2f b  = *(const v2f*)&Xs[lo][kr];
            const v2f cl = *(const v2f*)&crow[kr - 128];
            accA = wmma4(aF[k], b, accA);
            accC = wmma4(cl, b, accC);
        }
        // Elementwise update (read-phase for LDS)
        float xn[8];
#pragma unroll
        for (int j = 0; j < 8; ++j) {
            const float xc = Xs[lo][rbase + j];
            const float yi = accC[j] + byF[j];
            xn[j] = xc + STEPF * (accA[j] + tanhf(yi));
        }
        __syncthreads();  // all waves done reading old X
#pragma unroll
        for (int j = 0; j < 8; ++j) Xs[lo][rbase + j] = xn[j];
        // Per-lane rows are contiguous -> 2x b128 stores
        float* p = outLane + (size_t)t * NN;
        *(float4*)(p)     = make_float4(xn[0], xn[1], xn[2], xn[3]);
        *(float4*)(p + 4) = make_float4(xn[4], xn[5], xn[6], xn[7]);
        __syncthreads();  // new X visible before next read phase
    }
}

extern "C" void kernel_launch(void* const* d_in, const int* in_sizes, int n_in,
                              void* d_out, int out_size, void* d_ws, size_t ws_size,
                              hipStream_t stream) {
    const float* X0  = (const float*)d_in[0];  // (BS, N)
    const float* MA  = (const float*)d_in[1];  // (N, N)
    const float* MW  = (const float*)d_in[2];  // (N, N)
    const float* bAz = (const float*)d_in[3];  // scalar
    const float* bWz = (const float*)d_in[4];  // scalar
    const float* by  = (const float*)d_in[5];  // (N, 1)
    float* out = (float*)d_out;                // (BS, TMAX, N)

    float* A = (float*)d_ws;                   // 256 KB
    float* C = A + NN * NN;                    // 256 KB

    lrnn_prep<<<NN, NN, 0, stream>>>(MA, MW, bAz, bWz, A, C);
    lrnn_main<<<16, 512, 0, stream>>>(X0, A, C, by, out);
}